// GraphSage_54958401520041
// MI455X (gfx1250) — compile-verified
//
#include <hip/hip_runtime.h>
#include <hip/hip_bf16.h>

typedef __attribute__((ext_vector_type(2))) float v2f;
typedef __attribute__((ext_vector_type(8))) float v8f;

#define DD 128
#define TWO_D 256

// ---------------------------------------------------------------------------
// Gather + mean aggregate + concat:  h[i] = [ Z[nodes[i]] , mean_s Z[neigh[i,s]] ]
// One wave (32 lanes) per node; each lane owns 4 contiguous floats (float4).
// ---------------------------------------------------------------------------
__global__ void sage_aggregate(const float* __restrict__ Zt,
                               const int* __restrict__ nodes,
                               const int* __restrict__ neigh,
                               int S, float invS,
                               float* __restrict__ hbuf, int B)
{
    const int wave = threadIdx.x >> 5;
    const int lane = threadIdx.x & 31;
    const int i = blockIdx.x * 8 + wave;
    if (i >= B) return;
    const int c = lane * 4;

    const int* nb = neigh + (size_t)i * S;
    float4 acc = make_float4(0.f, 0.f, 0.f, 0.f);
    for (int s = 0; s < S; ++s) {
        const float* row = Zt + (size_t)nb[s] * DD + c;
        if (s + 1 < S)  // pull the next random row toward L2 early
            __builtin_prefetch(Zt + (size_t)nb[s + 1] * DD + c, 0, 0);
        float4 v = *(const float4*)row;
        acc.x += v.x; acc.y += v.y; acc.z += v.z; acc.w += v.w;
    }
    acc.x *= invS; acc.y *= invS; acc.z *= invS; acc.w *= invS;

    float4 sv = *(const float4*)(Zt + (size_t)nodes[i] * DD + c);

    float* hrow = hbuf + (size_t)i * TWO_D;
    *(float4*)(hrow + c)      = sv;   // self half
    *(float4*)(hrow + DD + c) = acc;  // neighbor-mean half
}

// ---------------------------------------------------------------------------
// z_pre = sigmoid(h @ W + b):  h [B,256] f32, W [256,128] row-major, via
// V_WMMA_F32_16X16X4_F32. One wave per 16x16 output tile; 8 waves per block
// cover one 16-row slab across all 128 columns. K=256 stepped by 4.
// A frag: lane holds h[m0+lane%16][k + 2*(lane/16) + {0,1}]
// B frag: lane holds W[k + 2*(lane/16) + {0,1}][n0 + lane%16]
// C/D   : VGPR v, lane -> (M = m0 + v + 8*(lane/16), N = n0 + lane%16)
// ---------------------------------------------------------------------------
__global__ void sage_gemm_sigmoid(const float* __restrict__ hbuf,
                                  const float* __restrict__ W,
                                  const float* __restrict__ bias,
                                  float* __restrict__ zbuf)
{
    const int wave = threadIdx.x >> 5;  // n-tile 0..7
    const int lane = threadIdx.x & 31;
    const int m0 = blockIdx.x * 16;
    const int n0 = wave * 16;
    const int r = lane & 15;
    const int g = lane >> 4;            // K sub-group 0/1

    const float* arow = hbuf + (size_t)(m0 + r) * TWO_D + g * 2;
    const float* bcol = W + (size_t)(g * 2) * DD + n0 + r;

    v8f acc = {};
    #pragma unroll 4
    for (int k = 0; k < TWO_D; k += 4) {
        v2f a = *(const v2f*)(arow + k);
        v2f b;
        b.x = bcol[(size_t)k * DD];
        b.y = bcol[(size_t)k * DD + DD];
        acc = __builtin_amdgcn_wmma_f32_16x16x4_f32(
            /*neg_a=*/false, a, /*neg_b=*/false, b,
            /*c_mod=*/(short)0, acc, /*reuse_a=*/false, /*reuse_b=*/false);
    }

    const float bv = bias[n0 + r];
    #pragma unroll
    for (int v = 0; v < 8; ++v) {
        const int m = m0 + v + 8 * g;
        const float x = acc[v] + bv;
        const float s = 1.0f / (1.0f + __expf(-x));
        zbuf[(size_t)m * DD + n0 + r] = s;
    }
}

// ---------------------------------------------------------------------------
// Row-wise L2 normalize then scatter into the table: Zt[nodes[i]] = z[i]/||z[i]||
// One wave per row; wave32 xor-shuffle reduction for the squared norm.
// ---------------------------------------------------------------------------
__global__ void sage_norm_scatter(const float* __restrict__ zbuf,
                                  const int* __restrict__ nodes,
                                  float* __restrict__ Zt, int B)
{
    const int wave = threadIdx.x >> 5;
    const int lane = threadIdx.x & 31;
    const int i = blockIdx.x * 8 + wave;
    if (i >= B) return;
    const int c = lane * 4;

    float4 z = *(const float4*)(zbuf + (size_t)i * DD + c);
    float ss = z.x * z.x + z.y * z.y + z.z * z.z + z.w * z.w;
    #pragma unroll
    for (int off = 16; off > 0; off >>= 1)
        ss += __shfl_xor(ss, off, 32);
    const float inv = rsqrtf(ss);

    float4 o = make_float4(z.x * inv, z.y * inv, z.z * inv, z.w * inv);
    *(float4*)(Zt + (size_t)nodes[i] * DD + c) = o;
}

// ---------------------------------------------------------------------------
extern "C" void kernel_launch(void* const* d_in, const int* in_sizes, int n_in,
                              void* d_out, int out_size, void* d_ws, size_t ws_size,
                              hipStream_t stream)
{
    const float* Z      = (const float*)d_in[0];
    const float* W0     = (const float*)d_in[1];
    const float* b0     = (const float*)d_in[2];
    const float* W1     = (const float*)d_in[3];
    const float* b1     = (const float*)d_in[4];
    const int*   nodes  = (const int*)d_in[5];
    const int*   neigh0 = (const int*)d_in[6];
    const int*   neigh1 = (const int*)d_in[7];
    float* Zout = (float*)d_out;

    const int N  = in_sizes[0] / DD;    // 500000
    const int B  = in_sizes[5];         // 16384
    const int S0 = in_sizes[6] / B;     // 25
    const int S1 = in_sizes[7] / B;     // 10

    float* hbuf = (float*)d_ws;                       // B * 256 f32 = 16 MB
    float* zbuf = hbuf + (size_t)B * TWO_D;           // B * 128 f32 =  8 MB

    // Working copy of the embedding table (output == updated table).
    hipMemcpyAsync(Zout, Z, (size_t)N * DD * sizeof(float),
                   hipMemcpyDeviceToDevice, stream);

    dim3 blk(256);
    dim3 aggGrid((B + 7) / 8);
    dim3 gemmGrid(B / 16);

    // Layer 0 (S=25)
    sage_aggregate   <<<aggGrid,  blk, 0, stream>>>(Zout, nodes, neigh0, S0, 1.0f / (float)S0, hbuf, B);
    sage_gemm_sigmoid<<<gemmGrid, blk, 0, stream>>>(hbuf, W0, b0, zbuf);
    sage_norm_scatter<<<aggGrid,  blk, 0, stream>>>(zbuf, nodes, Zout, B);

    // Layer 1 (S=10) — reads the table updated by layer 0 (stream-ordered).
    sage_aggregate   <<<aggGrid,  blk, 0, stream>>>(Zout, nodes, neigh1, S1, 1.0f / (float)S1, hbuf, B);
    sage_gemm_sigmoid<<<gemmGrid, blk, 0, stream>>>(hbuf, W1, b1, zbuf);
    sage_norm_scatter<<<aggGrid,  blk, 0, stream>>>(zbuf, nodes, Zout, B);
}